// OutAggregate_23012434772430
// MI455X (gfx1250) — compile-verified
//
#include <hip/hip_runtime.h>

typedef __attribute__((ext_vector_type(2))) float v2f;
typedef __attribute__((ext_vector_type(8))) float v8f;

namespace {
constexpr int kB = 64;        // batches
constexpr int kQ = 900;       // queries
constexpr int kC = 91;        // logit classes
constexpr int kNW = 29;       // 29 * 32 = 928 bits >= 900
constexpr int kKPad = 928;    // padded K extent (= kNW*32) for B matrix
constexpr int kRowPad = 912;  // 57 * 16 padded rows for WMMA tiles
constexpr int kTiles = 57;
constexpr float kTB = 0.9f;
}

__global__ __launch_bounds__(1024)
void giou_suppress_agg(const float* __restrict__ bboxes, float* __restrict__ agg)
{
  // ~150 KB LDS (CDNA5: up to 320 KB per workgroup)
  __shared__ float s_x0[kQ], s_y0[kQ], s_x1[kQ], s_y1[kQ], s_ar[kQ];
  __shared__ float s_bext[kKPad * 8];        // (cx,cy,w,h,1,0,0,0) per row; rows>=900 zero
  __shared__ unsigned s_mask[kRowPad * kNW]; // row bitmasks
  __shared__ unsigned s_alive[kNW];
  __shared__ unsigned s_wordnz[kTiles];      // bit w: K-word w has any bit in this tile

  const int b = blockIdx.x;
  const int tid = threadIdx.x;
  const int nthr = blockDim.x;
  const float* bb = bboxes + (size_t)b * kQ * 4;

  // Phase 1: xyxy conversion + extended B matrix in LDS
  for (int i = tid; i < kQ; i += nthr) {
    float cx = bb[i * 4 + 0], cy = bb[i * 4 + 1];
    float w  = bb[i * 4 + 2], h  = bb[i * 4 + 3];
    float x0 = cx - 0.5f * w, y0 = cy - 0.5f * h;
    float x1 = cx + 0.5f * w, y1 = cy + 0.5f * h;
    s_x0[i] = x0; s_y0[i] = y0; s_x1[i] = x1; s_y1[i] = y1;
    s_ar[i] = (x1 - x0) * (y1 - y0);
  }
  for (int i = tid; i < kKPad * 8; i += nthr) {
    int j = i >> 3, c = i & 7;
    float v = 0.0f;
    if (j < kQ) {
      if (c < 4) v = bb[j * 4 + c];
      else if (c == 4) v = 1.0f;   // count column
    }
    s_bext[i] = v;
  }
  for (int i = tid; i < (kRowPad - kQ) * kNW; i += nthr)
    s_mask[kQ * kNW + i] = 0u;
  if (tid < kTiles) s_wordnz[tid] = 0u;
  __syncthreads();

  // Phase 2: thresholded pairwise GIoU -> bitmask (strict upper triangle).
  // giou > T  <=>  inter*ac - (ac-uni)*uni > T*uni*ac   (uni>0, ac>0; no divides)
  for (int job = tid; job < kQ * kNW; job += nthr) {
    int row = job / kNW;
    int w = job - row * kNW;
    float x0i = s_x0[row], y0i = s_y0[row], x1i = s_x1[row], y1i = s_y1[row];
    float ai = s_ar[row];
    unsigned bits = 0u;
    int j0 = w * 32;
#pragma unroll 8
    for (int t = 0; t < 32; ++t) {
      int j = j0 + t;
      int jc = (j < kQ) ? j : 0;
      float x0j = s_x0[jc], y0j = s_y0[jc], x1j = s_x1[jc], y1j = s_y1[jc];
      float aj = s_ar[jc];
      float iw = fminf(x1i, x1j) - fmaxf(x0i, x0j);
      float ih = fminf(y1i, y1j) - fmaxf(y0i, y0j);
      iw = fmaxf(iw, 0.0f);
      ih = fmaxf(ih, 0.0f);
      float inter = iw * ih;
      float uni = ai + aj - inter;
      float cw = fmaxf(x1i, x1j) - fminf(x0i, x0j);
      float ch = fmaxf(y1i, y1j) - fminf(y0i, y0j);
      float ac = cw * ch;
      bool cond = (inter * ac - (ac - uni) * uni) > (kTB * uni * ac);
      bool valid = (j < kQ) && (j > row) && cond;
      bits |= (valid ? 1u : 0u) << t;
    }
    s_mask[row * kNW + w] = bits;
  }
  __syncthreads();

  // Phase 3: sequential greedy suppression (wave 0; alive words held in registers).
  if (tid < 32) {
    unsigned aliveReg = 0xFFFFFFFFu;    // lane w holds alive word w (w < kNW used)
    for (int i = 0; i < kQ; ++i) {
      unsigned aw = (unsigned)__shfl((int)aliveReg, i >> 5);  // uniform -> scalar
      if ((aw >> (i & 31)) & 1u)
        aliveReg &= ~s_mask[i * kNW + tid];  // lanes 29-31 touch padded words: harmless
    }
    if (tid < kNW) s_alive[tid] = aliveReg;
  }
  __syncthreads();

  // Phase 4a: final mask = (row if alive else 0) | identity
  for (int row = tid; row < kQ; row += nthr) {
    unsigned alive = (s_alive[row >> 5] >> (row & 31)) & 1u;
    if (!alive) {
      for (int w = 0; w < kNW; ++w) s_mask[row * kNW + w] = 0u;
    }
    s_mask[row * kNW + (row >> 5)] |= 1u << (row & 31);
  }
  __syncthreads();

  // Phase 4b: per-tile nonzero-word bitmap (skip map for the WMMA K loop)
  for (int job = tid; job < kTiles * kNW; job += nthr) {
    int t = job / kNW;
    int w = job - t * kNW;
    const unsigned* p = &s_mask[(t * 16) * kNW + w];
    unsigned orv = 0u;
#pragma unroll
    for (int r = 0; r < 16; ++r) orv |= p[r * kNW];
    if (orv) atomicOr(&s_wordnz[t], 1u << w);
  }
  __syncthreads();

  // Phase 5: WMMA aggregation. D[16x16] += A[16x4](mask bits) * B[4x16]
  const int lane = tid & 31;
  const int wave = tid >> 5;
  const int lhi = lane >> 4;                // A/B: lanes 16-31 carry K pair +2
  const int n = lane & 15;                  // B/D column
  const int ncl = (n < 5) ? n : 5;          // columns >=5 read a stored zero
  for (int tIdx = wave; tIdx < kTiles; tIdx += (nthr >> 5)) {
    const int row0 = tIdx * 16;
    const int arow = row0 + (lane & 15);
    v8f acc = {};
    // uniform skip map -> SGPR; iterate only set bits with a scalar loop
    unsigned rem = (unsigned)__builtin_amdgcn_readfirstlane((int)s_wordnz[tIdx]);
    for (; rem; rem &= rem - 1u) {
      int word = (int)__builtin_ctz(rem);
      unsigned mw = s_mask[arow * kNW + word];
#pragma unroll
      for (int ks = 0; ks < 8; ++ks) {
        int kbit = ks * 4 + (lhi << 1);
        v2f a, bv;
        a.x = ((mw >> kbit) & 1u) ? 1.0f : 0.0f;
        a.y = ((mw >> (kbit + 1)) & 1u) ? 1.0f : 0.0f;
        const float* p = &s_bext[(word * 32 + kbit) * 8 + ncl];
        bv.x = p[0];      // row kk
        bv.y = p[8];      // row kk+1  (ds_load_2addr pair)
        acc = __builtin_amdgcn_wmma_f32_16x16x4_f32(
            false, a, false, bv, (short)0, acc, false, false);
      }
    }
    // D layout: VGPR r -> lanes 0-15: (row0+r, col lane); lanes 16-31: (row0+r+8, col-16)
#pragma unroll
    for (int r = 0; r < 8; ++r) {
      float val = acc[r];
      float cnt = __shfl(val, (lane & 16) + 4);   // broadcast count column
      int rowg = row0 + r + (lhi << 3);
      if (rowg < kQ && n < 4)
        agg[((size_t)b * kQ + rowg) * 4 + n] = val / cnt;
    }
  }
}

extern "C" void kernel_launch(void* const* d_in, const int* in_sizes, int n_in,
                              void* d_out, int out_size, void* d_ws, size_t ws_size,
                              hipStream_t stream)
{
  (void)in_sizes; (void)n_in; (void)out_size; (void)d_ws; (void)ws_size;
  const float* bboxes = (const float*)d_in[0];
  const float* logits = (const float*)d_in[1];
  float* agg = (float*)d_out;                                  // [64*900*4]
  float* out_logits = (float*)d_out + (size_t)kB * kQ * 4;     // [64*900*91]

  giou_suppress_agg<<<dim3(kB), dim3(1024), 0, stream>>>(bboxes, agg);
  hipMemcpyAsync(out_logits, logits, (size_t)kB * kQ * kC * sizeof(float),
                 hipMemcpyDeviceToDevice, stream);
}